// BipartiteSAGELayer_88527865905575
// MI455X (gfx1250) — compile-verified
//
#include <hip/hip_runtime.h>
#include <hip/hip_bf16.h>

// ---------------------------------------------------------------------------
// BipartiteSAGELayer for MI455X (gfx1250, wave32)
//   Phase A: edge scatter-mean via global_atomic_add_f32 (L2-resident targets)
//   Phase B: fused concat -> GEMM (v_wmma_f32_16x16x4_f32) -> bias -> relu -> LN
// ---------------------------------------------------------------------------

#define N_JOBS  100000
#define N_MACH  10000
#define D_IN    128      // per-side feature dim; concat dim = 256
#define D_CAT   256
#define D_OUT   128
#define LN_EPS  1e-5f

typedef __attribute__((ext_vector_type(2))) float v2f;
typedef __attribute__((ext_vector_type(8))) float v8f;

// ---------------------------------------------------------------------------
// Edge scatter: one wave32 per edge. Lane l owns floats [4l, 4l+4) of the
// 128-float row. Gathers are float4 (global_load_b128); scatters are native
// fp32 atomics (global_atomic_add_f32 via unsafeAtomicAdd).
// ---------------------------------------------------------------------------
__global__ __launch_bounds__(256)
void sage_scatter_kernel(const float* __restrict__ job_h,
                         const float* __restrict__ machine_h,
                         const int*   __restrict__ job_idx,
                         const int*   __restrict__ machine_idx,
                         float* __restrict__ job_agg,
                         float* __restrict__ machine_agg,
                         float* __restrict__ job_deg,
                         float* __restrict__ machine_deg,
                         int nedges) {
    const int lane = threadIdx.x & 31;
    const int edge = (blockIdx.x * blockDim.x + threadIdx.x) >> 5;
    if (edge >= nedges) return;

    const int j = job_idx[edge];
    const int m = machine_idx[edge];
    const int c = lane << 2;

    const float4 mh = *(const float4*)(machine_h + (size_t)m * D_IN + c);
    const float4 jh = *(const float4*)(job_h     + (size_t)j * D_IN + c);

    float* ja = job_agg     + (size_t)j * D_IN + c;
    float* ma = machine_agg + (size_t)m * D_IN + c;

    unsafeAtomicAdd(ja + 0, mh.x);
    unsafeAtomicAdd(ja + 1, mh.y);
    unsafeAtomicAdd(ja + 2, mh.z);
    unsafeAtomicAdd(ja + 3, mh.w);

    unsafeAtomicAdd(ma + 0, jh.x);
    unsafeAtomicAdd(ma + 1, jh.y);
    unsafeAtomicAdd(ma + 2, jh.z);
    unsafeAtomicAdd(ma + 3, jh.w);

    if (lane == 0) {
        unsafeAtomicAdd(&job_deg[j], 1.0f);
        unsafeAtomicAdd(&machine_deg[m], 1.0f);
    }
}

// ---------------------------------------------------------------------------
// Fused concat + GEMM + bias + relu + layernorm.
// Block = 256 threads = 8 wave32. Block owns rows [16*bid, 16*bid+16).
// Wave w computes output columns [16w, 16w+16) with 64 x WMMA_F32_16X16X4_F32.
// nrows must be a multiple of 16 (100000 and 10000 both are).
//
// Fragment layouts (ISA 7.12.2, f32 16x16x4):
//   A (16x4): lane<16 -> M=lane, VGPR{0,1}=K{0,1}; lane>=16 -> M=lane-16, K{2,3}
//   B (4x16): lane<16 -> N=lane, VGPR{0,1}=K{0,1}; lane>=16 -> N=lane-16, K{2,3}
//   C/D     : lane -> N=lane&15, VGPR v -> M = v + 8*(lane>>4)
// ---------------------------------------------------------------------------
#define XS_STRIDE 264   // 256 + 8 pad; row stride 1056 B keeps 16B alignment

__global__ __launch_bounds__(256, 2)
void sage_gemm_ln_kernel(const float* __restrict__ h,      // [nrows,128]
                         const float* __restrict__ agg,    // [nrows,128] summed
                         const float* __restrict__ deg,    // [nrows]
                         const float* __restrict__ W,      // [128,256] row-major
                         const float* __restrict__ bias,   // [128]
                         const float* __restrict__ ln_s,   // [128]
                         const float* __restrict__ ln_b,   // [128]
                         float* __restrict__ out,          // [nrows,128]
                         int nrows) {
    __shared__ float Xs[16][XS_STRIDE];   // concat(h, agg/deg) tile
    __shared__ float Outs[16][D_OUT];     // relu'd GEMM tile

    const int tid  = threadIdx.x;
    const int wave = tid >> 5;
    const int lane = tid & 31;
    const int gRow0 = blockIdx.x * 16;
    if (gRow0 >= nrows) return;

    // ---- Phase 1: stage X tile (16 x 256) into LDS -------------------------
    {
        const int r     = tid >> 4;          // 0..15
        const int cbase = (tid & 15) << 4;   // 0,16,...,240
        const int gr    = gRow0 + r;
        float scale = 1.0f;
        const float* src;
        if (cbase < D_IN) {
            src = h + (size_t)gr * D_IN + cbase;
        } else {
            src = agg + (size_t)gr * D_IN + (cbase - D_IN);
            scale = 1.0f / fmaxf(deg[gr], 1.0f);   // scatter-mean
        }
        #pragma unroll
        for (int i = 0; i < 4; ++i) {
            float4 v = ((const float4*)src)[i];
            float4 o;
            o.x = v.x * scale; o.y = v.y * scale;
            o.z = v.z * scale; o.w = v.w * scale;
            *(float4*)&Xs[r][cbase + 4 * i] = o;
        }
    }
    __syncthreads();

    // ---- Phase 2: K-loop of 64 WMMA f32 16x16x4 ----------------------------
    const int nbase = wave << 4;                 // this wave's column block
    const int mrow  = lane & 15;                 // A: M index / B: N index
    const int koff  = (lane >> 4) << 1;          // 0 or 2
    const float* wp = W + (size_t)(nbase + mrow) * D_CAT + koff;

    v8f acc = {};
    #pragma unroll 8
    for (int t = 0; t < 64; ++t) {
        v2f a = *(const v2f*)&Xs[mrow][t * 4 + koff];   // ds_load_b64
        v2f b = *(const v2f*)(wp + t * 4);              // global_load_b64 (L2 hit)
        acc = __builtin_amdgcn_wmma_f32_16x16x4_f32(
                  /*neg_a=*/false, a, /*neg_b=*/false, b,
                  /*c_mod=*/(short)0, acc,
                  /*reuse_a=*/false, /*reuse_b=*/false);
    }

    // ---- Phase 3: bias + relu into LDS -------------------------------------
    {
        const int col  = nbase + (lane & 15);
        const float bc = bias[col];
        const int rb   = (lane >> 4) << 3;       // 0 or 8
        #pragma unroll
        for (int v = 0; v < 8; ++v) {
            Outs[rb + v][col] = fmaxf(acc[v] + bc, 0.0f);
        }
    }
    __syncthreads();

    // ---- Phase 4: layernorm; wave w handles rows 2w, 2w+1 ------------------
    #pragma unroll
    for (int rr = 0; rr < 2; ++rr) {
        const int r = wave * 2 + rr;
        float vals[4];
        float s = 0.0f, sq = 0.0f;
        #pragma unroll
        for (int i = 0; i < 4; ++i) {
            float v = Outs[r][lane + 32 * i];
            vals[i] = v;
            s  += v;
            sq += v * v;
        }
        #pragma unroll
        for (int m = 16; m >= 1; m >>= 1) {
            s  += __shfl_xor(s,  m, 32);
            sq += __shfl_xor(sq, m, 32);
        }
        const float mean = s * (1.0f / 128.0f);
        const float var  = sq * (1.0f / 128.0f) - mean * mean;
        const float rstd = rsqrtf(var + LN_EPS);
        const size_t ob  = (size_t)(gRow0 + r) * D_OUT;
        #pragma unroll
        for (int i = 0; i < 4; ++i) {
            const int c = lane + 32 * i;
            out[ob + c] = (vals[i] - mean) * rstd * ln_s[c] + ln_b[c];
        }
    }
}

// ---------------------------------------------------------------------------
extern "C" void kernel_launch(void* const* d_in, const int* in_sizes, int n_in,
                              void* d_out, int out_size, void* d_ws, size_t ws_size,
                              hipStream_t stream) {
    const float* job_h       = (const float*)d_in[0];
    const float* machine_h   = (const float*)d_in[1];
    const float* W_job_w     = (const float*)d_in[2];
    const float* W_job_b     = (const float*)d_in[3];
    const float* W_mach_w    = (const float*)d_in[4];
    const float* W_mach_b    = (const float*)d_in[5];
    const float* ln_j_scale  = (const float*)d_in[6];
    const float* ln_j_bias   = (const float*)d_in[7];
    const float* ln_m_scale  = (const float*)d_in[8];
    const float* ln_m_bias   = (const float*)d_in[9];
    const int*   job_idx     = (const int*)d_in[10];
    const int*   machine_idx = (const int*)d_in[11];
    float* out = (float*)d_out;

    // Workspace layout (floats):
    //   job_agg[N_JOBS*128] | machine_agg[N_MACH*128] | job_deg[N_JOBS] | machine_deg[N_MACH]
    float* job_agg     = (float*)d_ws;
    float* machine_agg = job_agg     + (size_t)N_JOBS * D_IN;
    float* job_deg     = machine_agg + (size_t)N_MACH * D_IN;
    float* machine_deg = job_deg     + N_JOBS;
    const size_t ws_floats = (size_t)N_JOBS * D_IN + (size_t)N_MACH * D_IN
                           + N_JOBS + N_MACH;

    hipMemsetAsync(d_ws, 0, ws_floats * sizeof(float), stream);

    const int nedges = in_sizes[10];
    // 8 wave32 per 256-thread block -> 8 edges per block
    const int scat_blocks = (nedges + 7) / 8;
    sage_scatter_kernel<<<scat_blocks, 256, 0, stream>>>(
        job_h, machine_h, job_idx, machine_idx,
        job_agg, machine_agg, job_deg, machine_deg, nedges);

    sage_gemm_ln_kernel<<<N_JOBS / 16, 256, 0, stream>>>(
        job_h, job_agg, job_deg, W_job_w, W_job_b,
        ln_j_scale, ln_j_bias, out, N_JOBS);

    sage_gemm_ln_kernel<<<N_MACH / 16, 256, 0, stream>>>(
        machine_h, machine_agg, machine_deg, W_mach_w, W_mach_b,
        ln_m_scale, ln_m_bias, out + (size_t)N_JOBS * D_OUT, N_MACH);
}